// GraphSAGE_TextMF_dyn_87411174408700
// MI455X (gfx1250) — compile-verified
//
#include <hip/hip_runtime.h>
#include <hip/hip_bf16.h>

typedef float v2f __attribute__((ext_vector_type(2)));
typedef float v8f __attribute__((ext_vector_type(8)));

#define D0     232
#define MED    1024
#define HID    256
#define TD     768
#define NTRAIN 4096
#define BATCH  16384
#define NEDGE  131072

#define MT 2   // 16-row M subtiles per wave
#define NT 4   // 16-col N subtiles per wave

// ---------------------------------------------------------------------------
// f32 WMMA GEMM:  C[M,N] = act( A1[M,K1] @ W1[N,K1]^T
//                               (+ A2[M,K2] @ W2[N,K2]^T)  + bias )
// One wave computes a 32x64 tile: MT*NT = 8 accumulators of
// V_WMMA_F32_16X16X4_F32. Per k-step: MT a-frag loads + NT b-frag loads +
// MT*NT WMMAs, no masking in the loop.
//
// N tail handling: WMMA columns are independent (lane l owns column n in both
// the B fragment and the C/D fragment), so out-of-range columns just read a
// clamped (valid, finite) weight row and their garbage results are never
// stored. EXEC stays all-ones through every WMMA.
// Call-site invariants: M % 32 == 0, K % 4 == 0.
// ---------------------------------------------------------------------------
__global__ __launch_bounds__(32)
void wmma_gemm_f32(const float* __restrict__ A1, const float* __restrict__ W1,
                   const float* __restrict__ A2, const float* __restrict__ W2,
                   const float* __restrict__ bias, const int* __restrict__ rowidx,
                   float* __restrict__ C,
                   int M, int N, int K1, int K2, int do_relu)
{
    const int lane = threadIdx.x & 31;
    const int l    = lane & 15;      // M index (A) / N index (B/C) within tile
    const int half = lane >> 4;      // selects K pair {0,1} vs {2,3}
    const int m0   = blockIdx.x * (16 * MT);
    const int n0   = blockIdx.y * (16 * NT);

    int arow[MT];
    #pragma unroll
    for (int mt = 0; mt < MT; ++mt) {
        int r = m0 + 16 * mt + l;
        arow[mt] = rowidx ? rowidx[r] : r;
    }

    int  nn[MT > NT ? MT : NT];      // (sized NT)
    bool nv[NT];
    int  ncl[NT];
    #pragma unroll
    for (int t = 0; t < NT; ++t) {
        nn[t]  = n0 + 16 * t + l;
        nv[t]  = nn[t] < N;
        ncl[t] = nv[t] ? nn[t] : (N - 1);   // clamped: loads stay in-bounds
    }

    v8f acc[MT][NT];
    #pragma unroll
    for (int mt = 0; mt < MT; ++mt)
        #pragma unroll
        for (int t = 0; t < NT; ++t)
            acc[mt][t] = (v8f){};

    // ---- pass 1: A1 @ W1^T ----
    {
        const float* Ar[MT];
        const float* Wr[NT];
        #pragma unroll
        for (int mt = 0; mt < MT; ++mt) Ar[mt] = A1 + (long)arow[mt] * K1;
        #pragma unroll
        for (int t = 0; t < NT; ++t)    Wr[t]  = W1 + (long)ncl[t]   * K1;

        for (int k0 = 0; k0 < K1; k0 += 4) {
            const int kk = k0 + 2 * half;
            v2f a[MT], b[NT];
            #pragma unroll
            for (int mt = 0; mt < MT; ++mt) a[mt] = *(const v2f*)(Ar[mt] + kk);
            #pragma unroll
            for (int t = 0; t < NT; ++t)    b[t]  = *(const v2f*)(Wr[t] + kk);
            #pragma unroll
            for (int mt = 0; mt < MT; ++mt)
                #pragma unroll
                for (int t = 0; t < NT; ++t)
                    acc[mt][t] = __builtin_amdgcn_wmma_f32_16x16x4_f32(
                        false, a[mt], false, b[t], (short)0, acc[mt][t], false, false);
        }
    }

    // ---- pass 2 (optional): + A2 @ W2^T ----
    if (A2) {
        const float* Ar[MT];
        const float* Wr[NT];
        #pragma unroll
        for (int mt = 0; mt < MT; ++mt) Ar[mt] = A2 + (long)(m0 + 16 * mt + l) * K2;
        #pragma unroll
        for (int t = 0; t < NT; ++t)    Wr[t]  = W2 + (long)ncl[t] * K2;

        for (int k0 = 0; k0 < K2; k0 += 4) {
            const int kk = k0 + 2 * half;
            v2f a[MT], b[NT];
            #pragma unroll
            for (int mt = 0; mt < MT; ++mt) a[mt] = *(const v2f*)(Ar[mt] + kk);
            #pragma unroll
            for (int t = 0; t < NT; ++t)    b[t]  = *(const v2f*)(Wr[t] + kk);
            #pragma unroll
            for (int mt = 0; mt < MT; ++mt)
                #pragma unroll
                for (int t = 0; t < NT; ++t)
                    acc[mt][t] = __builtin_amdgcn_wmma_f32_16x16x4_f32(
                        false, a[mt], false, b[t], (short)0, acc[mt][t], false, false);
        }
    }

    // ---- store: C/D layout -> VGPR v, lanes 0-15: M=v, lanes 16-31: M=v+8 ----
    #pragma unroll
    for (int t = 0; t < NT; ++t) {
        if (!nv[t]) continue;                       // divergence after all WMMAs
        const float bv = bias ? bias[nn[t]] : 0.f;
        #pragma unroll
        for (int mt = 0; mt < MT; ++mt) {
            const long mbase = (long)(m0 + 16 * mt + 8 * half);
            #pragma unroll
            for (int v = 0; v < 8; ++v) {
                float val = acc[mt][t][v] + bv;
                if (do_relu) val = fmaxf(val, 0.f);
                C[(mbase + v) * N + nn[t]] = val;
            }
        }
    }
}

// ---------------------------------------------------------------------------
// Graph mean-aggregation pieces (segment_sum semantics via f32 atomics)
// ---------------------------------------------------------------------------
__global__ __launch_bounds__(256)
void deg_kernel(const int* __restrict__ dst, float* __restrict__ deg, int E)
{
    int e = blockIdx.x * blockDim.x + threadIdx.x;
    if (e < E) atomicAdd(&deg[dst[e]], 1.0f);
}

__global__ __launch_bounds__(256)
void scatter_add_kernel(const float* __restrict__ x, const int* __restrict__ src,
                        const int* __restrict__ dst, float* __restrict__ agg,
                        int E, int D)
{
    long t = (long)blockIdx.x * blockDim.x + threadIdx.x;
    if (t >= (long)E * D) return;
    int e = (int)(t / D);
    int d = (int)(t % D);
    atomicAdd(&agg[(long)dst[e] * D + d], x[(long)src[e] * D + d]);
}

__global__ __launch_bounds__(256)
void scale_by_deg_kernel(float* __restrict__ agg, const float* __restrict__ deg,
                         int Nn, int D)
{
    long t = (long)blockIdx.x * blockDim.x + threadIdx.x;
    if (t >= (long)Nn * D) return;
    int i = (int)(t / D);
    agg[t] = agg[t] / fmaxf(deg[i], 1.0f);
}

// ---------------------------------------------------------------------------
// Final: out[i,c] = sum_d (x2[model_ids[i],d] * q[i,d]) * clsW[c,d] + clsb[c]
// ---------------------------------------------------------------------------
__global__ __launch_bounds__(256)
void final_cls_kernel(const float* __restrict__ x2, const float* __restrict__ q,
                      const int* __restrict__ model_ids,
                      const float* __restrict__ clsW, const float* __restrict__ clsb,
                      float* __restrict__ out, int Bn)
{
    int i = blockIdx.x * blockDim.x + threadIdx.x;
    if (i >= Bn) return;
    const float* p  = x2 + (long)model_ids[i] * D0;
    const float* qi = q  + (long)i * D0;
    float a0 = 0.f, a1 = 0.f;
    for (int d = 0; d < D0; ++d) {
        float pq = p[d] * qi[d];
        a0 += pq * clsW[d];
        a1 += pq * clsW[D0 + d];
    }
    out[i * 2 + 0] = a0 + clsb[0];
    out[i * 2 + 1] = a1 + clsb[1];
}

// ---------------------------------------------------------------------------
extern "C" void kernel_launch(void* const* d_in, const int* in_sizes, int n_in,
                              void* d_out, int out_size, void* d_ws, size_t ws_size,
                              hipStream_t stream)
{
    const float* P     = (const float*)d_in[0];   // (5000,1024) — rows [0,4096) used
    const float* Wp    = (const float*)d_in[1];   // (232,1024)
    const float* bp    = (const float*)d_in[2];   // (232)
    const float* s1Wl  = (const float*)d_in[3];   // (256,232)
    const float* s1bl  = (const float*)d_in[4];   // (256)
    const float* s1Wr  = (const float*)d_in[5];   // (256,232)
    const float* s2Wl  = (const float*)d_in[6];   // (232,256)
    const float* s2bl  = (const float*)d_in[7];   // (232)
    const float* s2Wr  = (const float*)d_in[8];   // (232,256)
    const float* Q     = (const float*)d_in[9];   // (100000,768)
    const float* Wt    = (const float*)d_in[10];  // (232,768)
    const float* bt    = (const float*)d_in[11];  // (232)
    const float* clsW  = (const float*)d_in[12];  // (2,232)
    const float* clsb  = (const float*)d_in[13];  // (2)
    const int*   ei    = (const int*)d_in[14];    // (2,131072): src then dst
    const int*   mids  = (const int*)d_in[15];    // (16384)
    const int*   pids  = (const int*)d_in[16];    // (16384)
    float* out = (float*)d_out;

    // Workspace layout (floats)
    float* x0   = (float*)d_ws;                 // 4096*232
    float* agg1 = x0   + (long)NTRAIN * D0;     // 4096*232
    float* x1   = agg1 + (long)NTRAIN * D0;     // 4096*256
    float* agg2 = x1   + (long)NTRAIN * HID;    // 4096*256
    float* x2   = agg2 + (long)NTRAIN * HID;    // 4096*232
    float* qbuf = x2   + (long)NTRAIN * D0;     // 16384*232
    float* deg  = qbuf + (long)BATCH  * D0;     // 4096

    const int* esrc = ei;
    const int* edst = ei + NEDGE;

    // zero atomic targets (capture-safe)
    hipMemsetAsync(deg,  0, NTRAIN * sizeof(float), stream);
    hipMemsetAsync(agg1, 0, (long)NTRAIN * D0  * sizeof(float), stream);
    hipMemsetAsync(agg2, 0, (long)NTRAIN * HID * sizeof(float), stream);

    const int MTILE = 16 * MT;   // 32
    const int NTILE = 16 * NT;   // 64

    // (1) x0 = P[:4096] @ Wp^T + bp                      (4096,232)
    wmma_gemm_f32<<<dim3(NTRAIN / MTILE, (D0 + NTILE - 1) / NTILE), 32, 0, stream>>>(
        P, Wp, nullptr, nullptr, bp, nullptr, x0, NTRAIN, D0, MED, 0, 0);

    // (2) degree + mean aggregation of x0
    deg_kernel<<<(NEDGE + 255) / 256, 256, 0, stream>>>(edst, deg, NEDGE);
    {
        long tot = (long)NEDGE * D0;
        scatter_add_kernel<<<(tot + 255) / 256, 256, 0, stream>>>(x0, esrc, edst, agg1, NEDGE, D0);
        long sc = (long)NTRAIN * D0;
        scale_by_deg_kernel<<<(sc + 255) / 256, 256, 0, stream>>>(agg1, deg, NTRAIN, D0);
    }

    // (3) x1 = relu(agg1 @ s1Wl^T + s1bl + x0 @ s1Wr^T)  (4096,256)
    wmma_gemm_f32<<<dim3(NTRAIN / MTILE, (HID + NTILE - 1) / NTILE), 32, 0, stream>>>(
        agg1, s1Wl, x0, s1Wr, s1bl, nullptr, x1, NTRAIN, HID, D0, D0, 1);

    // (4) mean aggregation of x1
    {
        long tot = (long)NEDGE * HID;
        scatter_add_kernel<<<(tot + 255) / 256, 256, 0, stream>>>(x1, esrc, edst, agg2, NEDGE, HID);
        long sc = (long)NTRAIN * HID;
        scale_by_deg_kernel<<<(sc + 255) / 256, 256, 0, stream>>>(agg2, deg, NTRAIN, HID);
    }

    // (5) x2 = agg2 @ s2Wl^T + s2bl + x1 @ s2Wr^T        (4096,232)
    wmma_gemm_f32<<<dim3(NTRAIN / MTILE, (D0 + NTILE - 1) / NTILE), 32, 0, stream>>>(
        agg2, s2Wl, x1, s2Wr, s2bl, nullptr, x2, NTRAIN, D0, HID, HID, 0);

    // (6) q = Q[prompt_ids] @ Wt^T + bt                  (16384,232)
    wmma_gemm_f32<<<dim3(BATCH / MTILE, (D0 + NTILE - 1) / NTILE), 32, 0, stream>>>(
        Q, Wt, nullptr, nullptr, bt, pids, qbuf, BATCH, D0, TD, 0, 0);

    // (7) out = (x2[model_ids] * q) @ clsW^T + clsb      (16384,2)
    final_cls_kernel<<<(BATCH + 255) / 256, 256, 0, stream>>>(
        x2, qbuf, mids, clsW, clsb, out, BATCH);
}